// NaiveKHopGraphAttention_8143257994118
// MI455X (gfx1250) — compile-verified
//
#include <hip/hip_runtime.h>
#include <hip/hip_bf16.h>

#define NN 50000
#define EE 800000
#define DD 128
#define HH 8
#define HD 16
#define EPSV 1e-5f

typedef __attribute__((ext_vector_type(2))) float v2f;
typedef __attribute__((ext_vector_type(8))) float v8f;

// ---------------- wave32 reduction ----------------
__device__ __forceinline__ float wave_sum(float v) {
#pragma unroll
  for (int m = 16; m > 0; m >>= 1) v += __shfl_xor(v, m, 32);
  return v;
}

// float atomic max via int/uint monotonic mapping (init to -inf)
__device__ __forceinline__ void atomic_max_f32(float* addr, float val) {
  int iv = __float_as_int(val);
  if (iv >= 0) atomicMax((int*)addr, iv);
  else atomicMin((unsigned int*)addr, (unsigned int)iv);
}

// ---------------- init workspace ----------------
__global__ __launch_bounds__(256) void init_k(float* __restrict__ acc,
                                              float* __restrict__ seg_max,
                                              float* __restrict__ seg_sum) {
  int i = blockIdx.x * 256 + threadIdx.x;
  if (i < NN * DD) acc[i] = 0.0f;
  if (i < NN * HH) {
    seg_max[i] = -__builtin_huge_valf();
    seg_sum[i] = 0.0f;
  }
}

// ---------------- C = A @ W^T + bias   (A: Nx128 row-major, W: 128x128 row-major)
// One block = 16-row tile; 8 waves, wave w owns 16-col tile w. K swept in steps
// of 4 with V_WMMA_F32_16X16X4_F32 (full fp32 precision).
// A-frag (16x4 f32): lanes 0-15 hold K=k,k+1 of row M=lane; lanes 16-31 K=k+2,k+3.
// B-frag mirrors it on W rows (B^T[n][k] = W[n][k]).
__global__ __launch_bounds__(256) void gemm_nt_bias(const float* __restrict__ A,
                                                    const float* __restrict__ W,
                                                    const float* __restrict__ bias,
                                                    float* __restrict__ C) {
  const int m0   = blockIdx.x * 16;
  const int wv   = threadIdx.x >> 5;
  const int lane = threadIdx.x & 31;
  const int n0   = wv * 16;
  const int kk   = (lane >> 4) << 1;                 // 0 or 2
  const float* ap = A + (size_t)(m0 + (lane & 15)) * DD + kk;
  const float* wp = W + (size_t)(n0 + (lane & 15)) * DD + kk;

  v8f c = {};
#pragma unroll
  for (int k0 = 0; k0 < DD; k0 += 4) {
    v2f a = *(const v2f*)(ap + k0);
    v2f b = *(const v2f*)(wp + k0);
    c = __builtin_amdgcn_wmma_f32_16x16x4_f32(false, a, false, b, (short)0, c,
                                              false, false);
  }
  // C/D layout: VGPR v -> (M=v, N=lane) for lanes 0-15, (M=v+8, N=lane-16) for 16-31
  const int col   = n0 + (lane & 15);
  const int rbase = m0 + ((lane >> 4) << 3);
  const float bb  = bias[col];
#pragma unroll
  for (int v = 0; v < 8; ++v)
    C[(size_t)(rbase + v) * DD + col] = c[v] + bb;
}

// ---------------- edge pass 1: per-(edge,head) score + segment max ----------------
__global__ __launch_bounds__(256) void edge_scores(const int* __restrict__ aw,
                                                   const float* __restrict__ QX,
                                                   const float* __restrict__ KX,
                                                   float* __restrict__ scores,
                                                   float* __restrict__ seg_max) {
  long idx = (long)blockIdx.x * 256 + threadIdx.x;
  if (idx >= (long)EE * HH) return;
  int e = (int)(idx >> 3), h = (int)(idx & 7);
  int s = aw[e], d = aw[EE + e];
  const float4* q = (const float4*)(QX + (size_t)s * DD + h * HD);
  const float4* k = (const float4*)(KX + (size_t)d * DD + h * HD);
  float acc = 0.0f;
#pragma unroll
  for (int i = 0; i < 4; ++i) {
    float4 qa = q[i], ka = k[i];
    acc += qa.x * ka.x + qa.y * ka.y + qa.z * ka.z + qa.w * ka.w;
  }
  float sc = acc * 0.25f;  // 1/sqrt(16)
  scores[idx] = sc;
  atomic_max_f32(&seg_max[(size_t)s * HH + h], sc);
}

// ---------------- edge pass 2: exp, seg_sum, unnormalized weighted-V scatter ----
__global__ __launch_bounds__(256) void edge_accum(const int* __restrict__ aw,
                                                  const float* __restrict__ VX,
                                                  const float* __restrict__ scores,
                                                  const float* __restrict__ seg_max,
                                                  float* __restrict__ seg_sum,
                                                  float* __restrict__ acc) {
  long idx = (long)blockIdx.x * 256 + threadIdx.x;
  if (idx >= (long)EE * HH) return;
  int e = (int)(idx >> 3), h = (int)(idx & 7);
  int s = aw[e], d = aw[EE + e];
  float ex = expf(scores[idx] - seg_max[(size_t)s * HH + h]);
  unsafeAtomicAdd(&seg_sum[(size_t)s * HH + h], ex);
  const float4* v = (const float4*)(VX + (size_t)d * DD + h * HD);
  float* o = acc + (size_t)s * DD + h * HD;
#pragma unroll
  for (int i = 0; i < 4; ++i) {
    float4 vv = v[i];
    unsafeAtomicAdd(o + i * 4 + 0, ex * vv.x);
    unsafeAtomicAdd(o + i * 4 + 1, ex * vv.y);
    unsafeAtomicAdd(o + i * 4 + 2, ex * vv.z);
    unsafeAtomicAdd(o + i * 4 + 3, ex * vv.w);
  }
}

// ---------------- normalize by seg_sum + LayerNorm1 (one wave per node) --------
__global__ __launch_bounds__(256) void ln1_k(const float* __restrict__ acc,
                                             const float* __restrict__ seg_sum,
                                             const float* __restrict__ g,
                                             const float* __restrict__ b,
                                             float* __restrict__ out) {
  int node = (blockIdx.x * 256 + threadIdx.x) >> 5;
  int lane = threadIdx.x & 31;
  if (node >= NN) return;
  int d0 = lane * 4;
  float den = seg_sum[(size_t)node * HH + (d0 >> 4)];
  float inv = den > 0.0f ? 1.0f / den : 0.0f;  // isolated node -> zero row
  float4 a = *(const float4*)(acc + (size_t)node * DD + d0);
  float v0 = a.x * inv, v1 = a.y * inv, v2 = a.z * inv, v3 = a.w * inv;
  float mu = wave_sum(v0 + v1 + v2 + v3) * (1.0f / DD);
  float t0 = v0 - mu, t1 = v1 - mu, t2 = v2 - mu, t3 = v3 - mu;
  float var = wave_sum(t0 * t0 + t1 * t1 + t2 * t2 + t3 * t3) * (1.0f / DD);
  float rs = rsqrtf(var + EPSV);
  float* o = out + (size_t)node * DD + d0;
  o[0] = t0 * rs * g[d0 + 0] + b[d0 + 0];
  o[1] = t1 * rs * g[d0 + 1] + b[d0 + 1];
  o[2] = t2 * rs * g[d0 + 2] + b[d0 + 2];
  o[3] = t3 * rs * g[d0 + 3] + b[d0 + 3];
}

// ---------------- LayerNorm2 (one wave per node) ----------------
__global__ __launch_bounds__(256) void ln2_k(const float* __restrict__ Z,
                                             const float* __restrict__ g,
                                             const float* __restrict__ b,
                                             float* __restrict__ out) {
  int node = (blockIdx.x * 256 + threadIdx.x) >> 5;
  int lane = threadIdx.x & 31;
  if (node >= NN) return;
  int d0 = lane * 4;
  float4 a = *(const float4*)(Z + (size_t)node * DD + d0);
  float mu = wave_sum(a.x + a.y + a.z + a.w) * (1.0f / DD);
  float t0 = a.x - mu, t1 = a.y - mu, t2 = a.z - mu, t3 = a.w - mu;
  float var = wave_sum(t0 * t0 + t1 * t1 + t2 * t2 + t3 * t3) * (1.0f / DD);
  float rs = rsqrtf(var + EPSV);
  float* o = out + (size_t)node * DD + d0;
  o[0] = t0 * rs * g[d0 + 0] + b[d0 + 0];
  o[1] = t1 * rs * g[d0 + 1] + b[d0 + 1];
  o[2] = t2 * rs * g[d0 + 2] + b[d0 + 2];
  o[3] = t3 * rs * g[d0 + 3] + b[d0 + 3];
}

extern "C" void kernel_launch(void* const* d_in, const int* in_sizes, int n_in,
                              void* d_out, int out_size, void* d_ws, size_t ws_size,
                              hipStream_t stream) {
  const float* X  = (const float*)d_in[0];
  const int*   aw = (const int*)d_in[1];
  const float* Wq = (const float*)d_in[2];
  const float* bq = (const float*)d_in[3];
  const float* Wk = (const float*)d_in[4];
  const float* bk = (const float*)d_in[5];
  const float* Wv = (const float*)d_in[6];
  const float* bv = (const float*)d_in[7];
  const float* Wo = (const float*)d_in[8];
  const float* bo = (const float*)d_in[9];
  const float* g1 = (const float*)d_in[10];
  const float* b1 = (const float*)d_in[11];
  const float* g2 = (const float*)d_in[12];
  const float* b2 = (const float*)d_in[13];
  float* out = (float*)d_out;

  float* ws = (float*)d_ws;
  const size_t ND = (size_t)NN * DD;          // 6.4M floats; also == E*H
  float* QX      = ws;                        // later reused as Z
  float* KX      = ws + ND;                   // later reused as Y1
  float* VX      = ws + 2 * ND;
  float* acc     = ws + 3 * ND;
  float* scores  = ws + 4 * ND;               // E*H elements
  float* seg_max = ws + 5 * ND;               // N*H
  float* seg_sum = seg_max + (size_t)NN * HH; // N*H
  float* Y1 = KX;
  float* Z  = QX;

  dim3 blk(256);
  init_k<<<(NN * DD + 255) / 256, blk, 0, stream>>>(acc, seg_max, seg_sum);

  gemm_nt_bias<<<NN / 16, blk, 0, stream>>>(X, Wq, bq, QX);
  gemm_nt_bias<<<NN / 16, blk, 0, stream>>>(X, Wk, bk, KX);
  gemm_nt_bias<<<NN / 16, blk, 0, stream>>>(X, Wv, bv, VX);

  int eblocks = (int)(((long)EE * HH + 255) / 256);
  edge_scores<<<eblocks, blk, 0, stream>>>(aw, QX, KX, scores, seg_max);
  edge_accum<<<eblocks, blk, 0, stream>>>(aw, VX, scores, seg_max, seg_sum, acc);

  ln1_k<<<NN / 8, blk, 0, stream>>>(acc, seg_sum, g1, b1, Y1);
  gemm_nt_bias<<<NN / 16, blk, 0, stream>>>(Y1, Wo, bo, Z);
  ln2_k<<<NN / 8, blk, 0, stream>>>(Z, g2, b2, out);
}